// Layer1_65558380806203
// MI455X (gfx1250) — compile-verified
//
#include <hip/hip_runtime.h>
#include <math.h>

#define MDIM 256
#define DDIM 768
#define THRESHF 0.3f
#define EPSF 1e-8f

typedef __attribute__((ext_vector_type(2))) float v2f;
typedef __attribute__((ext_vector_type(4))) float v4f;
typedef __attribute__((ext_vector_type(8))) float v8f;

// ---------------------------------------------------------------------------
// km_dot = mem @ mem.T via V_WMMA_F32_16X16X4_F32 (fp32-exact matrix path).
// One wave per 16x16 output tile; 768/4 = 192 WMMA ops per tile.
// A 16x4 layout: lane m=lane&15, half h=lane>>4 holds K=2h,2h+1 in a.x,a.y.
// B 4x16 mirrors it (B[k,n] = mem[col_n, k], same addressing as A).
// D layout: vgpr r holds D[M = r + 8h, N = lane&15].
// ---------------------------------------------------------------------------
__global__ void gemm_kernel(const float* __restrict__ mem, float* __restrict__ km) {
    const int wave = threadIdx.x >> 5;
    const int tile = blockIdx.x * 4 + wave;      // 0..255
    const int tr = (tile >> 4) << 4;             // tile row base
    const int tc = (tile & 15) << 4;             // tile col base
    const int lane = threadIdx.x & 31;
    const int h = lane >> 4;
    const int m = lane & 15;

    const float* __restrict__ Arow = mem + (tr + m) * DDIM + 2 * h;
    const float* __restrict__ Brow = mem + (tc + m) * DDIM + 2 * h;

    v8f acc = {};
    for (int k = 0; k < DDIM; k += 4) {
        v2f a = *(const v2f*)(Arow + k);
        v2f b = *(const v2f*)(Brow + k);
        acc = __builtin_amdgcn_wmma_f32_16x16x4_f32(
            /*neg_a=*/false, a, /*neg_b=*/false, b,
            /*c_mod=*/(short)0, acc, /*reuse_a=*/false, /*reuse_b=*/false);
    }
#pragma unroll
    for (int r = 0; r < 8; ++r)
        km[(tr + r + 8 * h) * MDIM + tc + m] = acc[r];
}

// ---------------------------------------------------------------------------
// x statistics: nx = max(||x||, eps), Sx = sum(x)
// ---------------------------------------------------------------------------
__global__ void xstats_kernel(const float* __restrict__ x, float* __restrict__ scal) {
    __shared__ float ssum[256], ssq[256];
    const int t = threadIdx.x;
    float s = 0.f, q = 0.f;
    for (int d = t; d < DDIM; d += 256) { float v = x[d]; s += v; q += v * v; }
    ssum[t] = s; ssq[t] = q;
    __syncthreads();
    for (int off = 128; off > 0; off >>= 1) {
        if (t < off) { ssum[t] += ssum[t + off]; ssq[t] += ssq[t + off]; }
        __syncthreads();
    }
    if (t == 0) {
        scal[0] = fmaxf(sqrtf(ssq[0]), EPSF);   // nx
        scal[1] = ssum[0];                       // Sx
    }
}

// ---------------------------------------------------------------------------
// Per-row stats of mem: nm (raw norm), Sm, kx_dot
// ---------------------------------------------------------------------------
__global__ void rowstats_kernel(const float* __restrict__ mem, const float* __restrict__ x,
                                float* __restrict__ nm, float* __restrict__ Sm,
                                float* __restrict__ kx) {
    __shared__ float ss[256], sq[256], sd[256];
    const int i = blockIdx.x, t = threadIdx.x;
    const float* __restrict__ row = mem + i * DDIM;
    float s = 0.f, q = 0.f, dd = 0.f;
    for (int d = t; d < DDIM; d += 256) {
        float v = row[d];
        s += v; q += v * v; dd += v * x[d];
    }
    ss[t] = s; sq[t] = q; sd[t] = dd;
    __syncthreads();
    for (int off = 128; off > 0; off >>= 1) {
        if (t < off) { ss[t] += ss[t + off]; sq[t] += sq[t + off]; sd[t] += sd[t + off]; }
        __syncthreads();
    }
    if (t == 0) {
        nm[i] = sqrtf(sq[0]);
        Sm[i] = ss[0];
        kx[i] = sd[0];
    }
}

// ---------------------------------------------------------------------------
// mx_cos, mask, cnt, A2
// ---------------------------------------------------------------------------
__global__ void scalars_kernel(const float* __restrict__ nm, const float* __restrict__ Sm,
                               const float* __restrict__ kx,
                               float* __restrict__ mxc, float* __restrict__ maskv,
                               float* __restrict__ scal) {
    __shared__ float scnt[256], sA2[256];
    const int i = threadIdx.x;
    const float nx = scal[0];
    const float nms = fmaxf(nm[i], EPSF);
    const float m = kx[i] / (nms * nx);
    const float mk = (m > THRESHF) ? 1.f : 0.f;
    mxc[i] = m;
    maskv[i] = mk;
    scnt[i] = mk;
    sA2[i] = mk * Sm[i] / nms;
    __syncthreads();
    for (int off = 128; off > 0; off >>= 1) {
        if (i < off) { scnt[i] += scnt[i + off]; sA2[i] += sA2[i + off]; }
        __syncthreads();
    }
    if (i == 0) { scal[2] = scnt[0]; scal[3] = sA2[0]; }
}

// ---------------------------------------------------------------------------
// A1[i] = sum_j km_dot[i,j] * mask[j] / nm_s[j]
// ---------------------------------------------------------------------------
__global__ void a1_kernel(const float* __restrict__ km, const float* __restrict__ maskv,
                          const float* __restrict__ nm, float* __restrict__ A1v) {
    __shared__ float sbuf[256];
    const int i = blockIdx.x, t = threadIdx.x;
    const float nms = fmaxf(nm[t], EPSF);
    sbuf[t] = km[i * MDIM + t] * maskv[t] / nms;
    __syncthreads();
    for (int off = 128; off > 0; off >>= 1) {
        if (t < off) sbuf[t] += sbuf[t + off];
        __syncthreads();
    }
    if (t == 0) A1v[i] = sbuf[0];
}

// ---------------------------------------------------------------------------
// out[a,i,:] = mask[a] ? (mem[i,:] + s[a,i] + noise[a,i,:]) : 0
// One wave32 per (a,i) pair; 6 coalesced float4 sweeps of 128 floats.
// noise is read-once and out is write-once: both use non-temporal cache
// policy (th:TH_NT) so the 400 MB stream does not evict the hot, 256x-reused
// mem rows (786 KB) from L2. Inactive rows skip the noise read entirely.
// ---------------------------------------------------------------------------
__global__ void out_kernel(const float* __restrict__ mem, const float* __restrict__ noise,
                           const float* __restrict__ km,
                           const float* __restrict__ nm, const float* __restrict__ Sm,
                           const float* __restrict__ kx, const float* __restrict__ mxc,
                           const float* __restrict__ maskv, const float* __restrict__ A1v,
                           const float* __restrict__ scal, float* __restrict__ out) {
    const int wave = threadIdx.x >> 5;
    const int lane = threadIdx.x & 31;
    const int p = blockIdx.x * 8 + wave;         // 0..65535
    const int a = p >> 8;
    const int i = p & 255;
    const int base = p * DDIM;                   // < 2^26, fits int

    if (maskv[a] == 0.f) {
        v4f z = {};
#pragma unroll
        for (int v = 0; v < 6; ++v)
            __builtin_nontemporal_store(z, (v4f*)(out + base + v * 128 + lane * 4));
        return;
    }

    const float nx = scal[0], Sx = scal[1], cnt = scal[2], A2 = scal[3];
    const float nmi = nm[i];
    const float nmsi = fmaxf(nmi, EPSF);
    const float nmsa = fmaxf(nm[a], EPSF);
    const float c = km[i * MDIM + a] / (nmsi * nmsa) + mxc[i];
    const float nkp = sqrtf(fmaxf(nmi * nmi + 2.f * c * Sm[i] + (float)DDIM * c * c,
                                  EPSF * EPSF));
    const float ckx = (kx[i] + c * Sx) / (nkp * nx);
    const float mean = (cnt > 0.f) ? (A1v[i] + c * A2) / (nkp * fmaxf(cnt, 1.f)) : 0.f;
    const float s = c + ckx + mean;

    const float* __restrict__ mrow = mem + i * DDIM;
    const float* __restrict__ nrow = noise + base;
    float* __restrict__ orow = out + base;
#pragma unroll
    for (int v = 0; v < 6; ++v) {
        const int off = v * 128 + lane * 4;
        v4f mv = *(const v4f*)(mrow + off);                                // RT: reused 256x
        v4f nv = __builtin_nontemporal_load((const v4f*)(nrow + off));     // NT: read once
        __builtin_nontemporal_store(mv + s + nv, (v4f*)(orow + off));      // NT: write once
    }
}

// ---------------------------------------------------------------------------
extern "C" void kernel_launch(void* const* d_in, const int* in_sizes, int n_in,
                              void* d_out, int out_size, void* d_ws, size_t ws_size,
                              hipStream_t stream) {
    const float* x     = (const float*)d_in[0];   // [1, 768]
    const float* mem   = (const float*)d_in[1];   // [256, 768]
    const float* noise = (const float*)d_in[2];   // [256, 256, 768]
    float* out = (float*)d_out;                   // [256, 256, 768]

    float* W     = (float*)d_ws;
    float* nm    = W;            // 256
    float* Sm    = W + 256;      // 256
    float* kx    = W + 512;      // 256
    float* mxc   = W + 768;      // 256
    float* maskv = W + 1024;     // 256
    float* A1v   = W + 1280;     // 256
    float* scal  = W + 1536;     // nx, Sx, cnt, A2
    float* km    = W + 1600;     // 256*256

    // Independent of row stats: launch the WMMA GEMM first for overlap.
    gemm_kernel<<<64, 128, 0, stream>>>(mem, km);
    xstats_kernel<<<1, 256, 0, stream>>>(x, scal);
    rowstats_kernel<<<256, 256, 0, stream>>>(mem, x, nm, Sm, kx);
    scalars_kernel<<<1, 256, 0, stream>>>(nm, Sm, kx, mxc, maskv, scal);
    a1_kernel<<<256, 256, 0, stream>>>(km, maskv, nm, A1v);
    out_kernel<<<8192, 256, 0, stream>>>(mem, noise, km, nm, Sm, kx, mxc, maskv,
                                         A1v, scal, out);
}